// GCN_27908697489547
// MI455X (gfx1250) — compile-verified
//
#include <hip/hip_runtime.h>
#include <hip/hip_bf16.h>

typedef __attribute__((ext_vector_type(2))) float v2f;
typedef __attribute__((ext_vector_type(8))) float v8f;

// ---------------------------------------------------------------------------
// C[M x 16] = A[M x K] @ W[K x 16], fp32 via V_WMMA_F32_16X16X4_F32.
// One wave32 per 16-row tile. K must be a multiple of 4.
// A is a single-pass stream -> non-temporal loads (TH=NT) so it does not
// evict the L2-resident hw/agg buffers (192MB L2; x alone is 204.8MB).
// ---------------------------------------------------------------------------
__global__ __launch_bounds__(256) void gcn_gemm_wmma_f32(
    const float* __restrict__ A, const float* __restrict__ W,
    float* __restrict__ C, int M, int K) {
  const int wave = blockIdx.x * (blockDim.x >> 5) + (threadIdx.x >> 5);
  const int lane = threadIdx.x & 31;
  const int m0 = wave << 4;
  if (m0 >= M) return;                 // whole wave exits together: EXEC stays all-1s
  const int half = lane >> 4;          // 0: lanes 0-15, 1: lanes 16-31
  const int l = lane & 15;

  v8f acc = {};
  const float* arow = A + (size_t)(m0 + l) * K + (half << 1); // K offset +0 or +2
  for (int k0 = 0; k0 < K; k0 += 4) {
    // A 16x4: VGPR v = A[m0+l][k0 + 2*half + v]  -> contiguous float2 per lane
    v2f a;
    a.x = __builtin_nontemporal_load(&arow[k0 + 0]);
    a.y = __builtin_nontemporal_load(&arow[k0 + 1]);
    // B 4x16: VGPR v = W[k0 + 2*half + v][l]   (16KB max, cache-hot: RT)
    v2f b;
    b.x = W[(size_t)(k0 + (half << 1) + 0) * 16 + l];
    b.y = W[(size_t)(k0 + (half << 1) + 1) * 16 + l];
    acc = __builtin_amdgcn_wmma_f32_16x16x4_f32(
        /*neg_a=*/false, a, /*neg_b=*/false, b,
        /*c_mod=*/(short)0, acc, /*reuse_a=*/false, /*reuse_b=*/false);
  }
  // C/D: VGPR r -> row m0 + r + 8*half, col l
  float* cp = C + ((size_t)(m0 + (half << 3)) << 4) + l;
#pragma unroll
  for (int r = 0; r < 8; ++r) cp[(size_t)r << 4] = acc[r];
}

// ---------------------------------------------------------------------------
// Degree / normalization helpers
// ---------------------------------------------------------------------------
__global__ void gcn_fill(float* __restrict__ p, float v, int n) {
  int i = blockIdx.x * blockDim.x + threadIdx.x;
  if (i < n) p[i] = v;
}

__global__ void gcn_deg_count(const int* __restrict__ dst, float* __restrict__ deg, int E) {
  int e = blockIdx.x * blockDim.x + threadIdx.x;
  if (e < E) atomicAdd(&deg[__builtin_nontemporal_load(&dst[e])], 1.0f);
}

__global__ void gcn_rsqrt(float* __restrict__ d, int n) {
  int i = blockIdx.x * blockDim.x + threadIdx.x;
  if (i < n) d[i] = rsqrtf(d[i]);
}

// ---------------------------------------------------------------------------
// agg[i][c] = dis[i]^2 * hw[i][c] + b[c]   (self-loop + bias initialization)
// ---------------------------------------------------------------------------
__global__ void gcn_agg_init(const float* __restrict__ hw, const float* __restrict__ dis,
                             const float* __restrict__ b, float* __restrict__ agg, int n16) {
  int t = blockIdx.x * blockDim.x + threadIdx.x;
  if (t >= n16) return;
  int i = t >> 4, c = t & 15;
  float d = dis[i];
  agg[t] = d * d * hw[t] + b[c];
}

// ---------------------------------------------------------------------------
// Edge scatter: 16 lanes per edge (one channel each).
// agg[dst][c] += dis[src]*dis[dst] * hw[src][c]
// hw/agg are 12.8 MB -> L2-resident; per-edge the 16 atomics hit one 64B line.
// Index reads are single-pass streams -> NT.
// ---------------------------------------------------------------------------
__global__ void gcn_edge_scatter16(const int* __restrict__ src, const int* __restrict__ dst,
                                   const float* __restrict__ dis, const float* __restrict__ hw,
                                   float* __restrict__ agg, int E) {
  int t = blockIdx.x * blockDim.x + threadIdx.x;
  int e = t >> 4, c = t & 15;
  if (e >= E) return;
  int s = __builtin_nontemporal_load(&src[e]);
  int d = __builtin_nontemporal_load(&dst[e]);
  float norm = dis[s] * dis[d];
  atomicAdd(&agg[((size_t)d << 4) + c], norm * hw[((size_t)s << 4) + c]);
}

__global__ void gcn_relu(float* __restrict__ p, int n) {
  int i = blockIdx.x * blockDim.x + threadIdx.x;
  if (i < n) p[i] = fmaxf(p[i], 0.0f);
}

// hw3[i] = dot(h[i,:16], W3[:,0]); h is last-use here -> NT reads
__global__ void gcn_mv16(const float* __restrict__ h, const float* __restrict__ W3,
                         float* __restrict__ out, int N) {
  int i = blockIdx.x * blockDim.x + threadIdx.x;
  if (i >= N) return;
  const float* hp = h + ((size_t)i << 4);
  float s = 0.f;
#pragma unroll
  for (int c = 0; c < 16; ++c) s += __builtin_nontemporal_load(&hp[c]) * W3[c];
  out[i] = s;
}

__global__ void gcn_out_init(const float* __restrict__ hw3, const float* __restrict__ dis,
                             const float* __restrict__ b3, float* __restrict__ out, int N) {
  int i = blockIdx.x * blockDim.x + threadIdx.x;
  if (i >= N) return;
  float d = dis[i];
  out[i] = d * d * hw3[i] + b3[0];
}

__global__ void gcn_edge_scatter1(const int* __restrict__ src, const int* __restrict__ dst,
                                  const float* __restrict__ dis, const float* __restrict__ hw3,
                                  float* __restrict__ out, int E) {
  int e = blockIdx.x * blockDim.x + threadIdx.x;
  if (e >= E) return;
  int s = __builtin_nontemporal_load(&src[e]);
  int d = __builtin_nontemporal_load(&dst[e]);
  atomicAdd(&out[d], dis[s] * dis[d] * hw3[s]);
}

// ---------------------------------------------------------------------------
static inline int cdiv(long long a, long long b) { return (int)((a + b - 1) / b); }

extern "C" void kernel_launch(void* const* d_in, const int* in_sizes, int n_in,
                              void* d_out, int out_size, void* d_ws, size_t ws_size,
                              hipStream_t stream) {
  const float* x  = (const float*)d_in[0];
  const int*   ei = (const int*)d_in[1];
  const float* W1 = (const float*)d_in[2];
  const float* b1 = (const float*)d_in[3];
  const float* W2 = (const float*)d_in[4];
  const float* b2 = (const float*)d_in[5];
  const float* W3 = (const float*)d_in[6];
  const float* b3 = (const float*)d_in[7];

  const int IN_CH = 256;
  const int HID = 16;
  const int N = in_sizes[0] / IN_CH;   // 200000
  const int E = in_sizes[1] / 2;       // 6400000
  const int* src = ei;
  const int* dst = ei + E;
  float* out = (float*)d_out;

  // workspace carving (256B aligned)
  char* ws = (char*)d_ws;
  size_t off = 0;
  float* dis = (float*)(ws + off); off += (((size_t)N * 4) + 255) & ~(size_t)255;
  float* HW  = (float*)(ws + off); off += (((size_t)N * HID * 4) + 255) & ~(size_t)255;
  float* AGG = (float*)(ws + off); off += (((size_t)N * HID * 4) + 255) & ~(size_t)255;
  (void)ws_size; (void)n_in; (void)out_size;

  const int B = 256;
  const int n16 = N * HID;             // 3.2M
  const int gemmBlocks = cdiv(cdiv(N, 16), B / 32);

  // degrees: deg = 1 (self loop) + incoming count; dis = deg^-1/2
  gcn_fill<<<cdiv(N, B), B, 0, stream>>>(dis, 1.0f, N);
  gcn_deg_count<<<cdiv(E, B), B, 0, stream>>>(dst, dis, E);
  gcn_rsqrt<<<cdiv(N, B), B, 0, stream>>>(dis, N);

  // ---- layer 1 ----
  gcn_gemm_wmma_f32<<<gemmBlocks, B, 0, stream>>>(x, W1, HW, N, IN_CH);
  gcn_agg_init<<<cdiv(n16, B), B, 0, stream>>>(HW, dis, b1, AGG, n16);
  gcn_edge_scatter16<<<cdiv((long long)E * 16, B), B, 0, stream>>>(src, dst, dis, HW, AGG, E);
  gcn_relu<<<cdiv(n16, B), B, 0, stream>>>(AGG, n16);        // AGG now holds h1

  // ---- layer 2 ----
  gcn_gemm_wmma_f32<<<gemmBlocks, B, 0, stream>>>(AGG, W2, HW, N, HID);
  gcn_agg_init<<<cdiv(n16, B), B, 0, stream>>>(HW, dis, b2, AGG, n16);
  gcn_edge_scatter16<<<cdiv((long long)E * 16, B), B, 0, stream>>>(src, dst, dis, HW, AGG, E);
  gcn_relu<<<cdiv(n16, B), B, 0, stream>>>(AGG, n16);        // AGG now holds h2

  // ---- layer 3 (16 -> 1) ----
  gcn_mv16<<<cdiv(N, B), B, 0, stream>>>(AGG, W3, HW, N);    // HW[0..N) = hw3
  gcn_out_init<<<cdiv(N, B), B, 0, stream>>>(HW, dis, b3, out, N);
  gcn_edge_scatter1<<<cdiv(E, B), B, 0, stream>>>(src, dst, dis, HW, out, E);
}